// AtLBase_54254026883782
// MI455X (gfx1250) — compile-verified
//
#include <hip/hip_runtime.h>

#define ANGLE_NUM 8500
#define MP 8512            // 8500 padded to multiple of 64 (133 * 64)
#define DMODEL 32
#define DFF 64
#define NLAYERS 8
#define MLP_CH 512
#define NKT (MP / 16)      // 532 key tiles

typedef __attribute__((ext_vector_type(2))) float v2f;
typedef __attribute__((ext_vector_type(8))) float v8f;
typedef int v4i __attribute__((vector_size(16)));

__device__ __forceinline__ v8f wmma_f32_k4(v2f a, v2f b, v8f c) {
  // D = A(16x4 f32) * B(4x16 f32) + C(16x16 f32)
  return __builtin_amdgcn_wmma_f32_16x16x4_f32(false, a, false, b, (short)0, c,
                                               false, false);
}

// ---------------------------------------------------------------------------
// CDNA5 async global->LDS copy (ASYNCcnt-tracked), with safe fallback.
#if defined(__gfx1250__) && __has_builtin(__builtin_amdgcn_global_load_async_to_lds_b128)
#define USE_ASYNC_LDS 1
#endif

__device__ __forceinline__ void lds_copy16(void* lds_dst, const void* g_src) {
#if defined(USE_ASYNC_LDS)
  __builtin_amdgcn_global_load_async_to_lds_b128(
      (__attribute__((address_space(1))) v4i*)g_src,
      (__attribute__((address_space(3))) v4i*)lds_dst, 0, 0);
#else
  *(float4*)lds_dst = *(const float4*)g_src;
#endif
}

__device__ __forceinline__ void async_wait0() {
#if defined(USE_ASYNC_LDS)
#if __has_builtin(__builtin_amdgcn_s_wait_asynccnt)
  __builtin_amdgcn_s_wait_asynccnt(0);
#else
  asm volatile("s_wait_asynccnt 0x0" ::: "memory");
#endif
#endif
}

__device__ __forceinline__ void async_wait2() {
#if defined(USE_ASYNC_LDS)
#if __has_builtin(__builtin_amdgcn_s_wait_asynccnt)
  __builtin_amdgcn_s_wait_asynccnt(2);
#else
  asm volatile("s_wait_asynccnt 0x2" ::: "memory");
#endif
#endif
}

// ---------------------------------------------------------------------------
// x[row,d] = embed[angle[row], d] * intensity[row]; pad rows -> 0
__global__ void embed_kernel(const float* __restrict__ intensity,
                             const int* __restrict__ angle,
                             const float* __restrict__ embed,
                             float* __restrict__ x) {
  int idx = blockIdx.x * blockDim.x + threadIdx.x;
  if (idx >= MP * DMODEL) return;
  int row = idx / DMODEL;
  int d = idx % DMODEL;
  float val = 0.f;
  if (row < ANGLE_NUM) val = embed[angle[row] * DMODEL + d] * intensity[row];
  x[idx] = val;
}

// ---------------------------------------------------------------------------
// Generic tiled GEMM: out[M,N] = A[M,K] @ W[K,N] + bias (+resid) (+leaky relu)
// 4 waves per block, one 16x16 output tile per wave.
// grid = (M/64, N/16), block = 128.
__global__ void gemm16_kernel(const float* __restrict__ A,
                              const float* __restrict__ W,
                              const float* __restrict__ bias,
                              const float* __restrict__ resid,
                              float* __restrict__ out,
                              int K, int N, int act) {
  const int lane = threadIdx.x & 31;
  const int wv = threadIdx.x >> 5;
  const int hf = lane >> 4;      // half-wave select
  const int col = lane & 15;     // M index for A-frag, N index for B/D-frag
  const int m0 = (blockIdx.x * 4 + wv) * 16;
  const int n0 = blockIdx.y * 16;

  v8f acc = {};
  const float* arow = A + (size_t)(m0 + col) * K;
  for (int kk = 0; kk < K; kk += 4) {
    int ka = kk + 2 * hf;
    v2f a;
    a[0] = arow[ka];
    a[1] = arow[ka + 1];
    v2f b;
    b[0] = W[(size_t)ka * N + n0 + col];
    b[1] = W[(size_t)(ka + 1) * N + n0 + col];
    acc = wmma_f32_k4(a, b, acc);
  }

  float bv = bias[n0 + col];
#pragma unroll
  for (int r = 0; r < 8; ++r) {
    int row = m0 + r + 8 * hf;
    float v = acc[r] + bv;
    if (resid) v += resid[(size_t)row * N + n0 + col];
    if (act) v = v > 0.f ? v : 0.01f * v;   // leaky relu, slope 0.01
    out[(size_t)row * N + n0 + col] = v;
  }
}

// ---------------------------------------------------------------------------
// Flash attention. Block = 128 threads = 4 waves; each wave owns a 16-row
// query tile (block covers 64 rows). K/V tiles (16x32) are staged into
// double-buffered LDS by async global->LDS copies shared by all 4 waves.
// O = softmax(Q K^T / sqrt(D)) V + intensity * V
__global__ void attn_kernel(const float* __restrict__ Q,
                            const float* __restrict__ Km,
                            const float* __restrict__ V,
                            const float* __restrict__ inten,
                            float* __restrict__ O) {
  __shared__ __align__(16) float sK[2][16][DMODEL];
  __shared__ __align__(16) float sV[2][16][DMODEL];
  __shared__ float sP[4][16][17];   // per-wave prob tile (17: bank pad)
  __shared__ float sStat[4][16];

  const int tid = threadIdx.x;
  const int lane = tid & 31;
  const int wv = tid >> 5;
  const int hf = lane >> 4;
  const int col = lane & 15;
  const int m0 = blockIdx.x * 64 + wv * 16;
  const float scale = 0.17677669529663687f;  // 1/sqrt(32)

  // Cooperative staging indices: 128 threads x 16B cover one 16x32 tile.
  const int lr = tid >> 3;          // 0..15 row
  const int lc = (tid & 7) * 4;     // 0,4,...,28 col

  v8f o0 = {}, o1 = {};
  float m_run = -1e30f, l_run = 0.f;

  // Preload Q A-frags for full K=32 (8 steps x v2f)
  v2f qa[8];
  {
    const float* qrow = Q + (size_t)(m0 + col) * DMODEL;
#pragma unroll
    for (int t = 0; t < 8; ++t) {
      int ka = 4 * t + 2 * hf;
      qa[t][0] = qrow[ka];
      qa[t][1] = qrow[ka + 1];
    }
  }

  // Prologue: stage tile 0 into buffer 0.
  lds_copy16(&sK[0][lr][lc], Km + (size_t)lr * DMODEL + lc);
  lds_copy16(&sV[0][lr][lc], V + (size_t)lr * DMODEL + lc);

  for (int kt = 0; kt < NKT; ++kt) {
    const int kbase = kt * 16;
    const int buf = kt & 1;

    if (kt + 1 < NKT) {
      const size_t nb = (size_t)(kbase + 16 + lr) * DMODEL + lc;
      lds_copy16(&sK[buf ^ 1][lr][lc], Km + nb);
      lds_copy16(&sV[buf ^ 1][lr][lc], V + nb);
      async_wait2();   // current tile's copies (issued 1 round ago) complete
    } else {
      async_wait0();
    }
    __syncthreads();   // staged tile visible to all 4 waves

    // S = Q @ K^T  (B-frag: B[k][n] = K[kbase+n][k], read from LDS)
    v8f s = {};
#pragma unroll
    for (int t = 0; t < 8; ++t) {
      int ka = 4 * t + 2 * hf;
      v2f b;
      b[0] = sK[buf][col][ka];
      b[1] = sK[buf][col][ka + 1];
      s = wmma_f32_k4(qa[t], b, s);
    }

#pragma unroll
    for (int r = 0; r < 8; ++r) sP[wv][r + 8 * hf][col] = s[r] * scale;
    __syncthreads();

    // Online softmax per row (lanes 0..15 of each wave own one row each);
    // mask padded keys (kbase+j >= ANGLE_NUM).
    if (lane < 16) {
      const int row = lane;
      float rmax = -1e30f;
#pragma unroll
      for (int j = 0; j < 16; ++j)
        if (kbase + j < ANGLE_NUM) rmax = fmaxf(rmax, sP[wv][row][j]);
      float mnew = fmaxf(m_run, rmax);
      float alpha = __expf(m_run - mnew);
      float lsum = 0.f;
#pragma unroll
      for (int j = 0; j < 16; ++j) {
        float p = 0.f;
        if (kbase + j < ANGLE_NUM) p = __expf(sP[wv][row][j] - mnew);
        sP[wv][row][j] = p;
        lsum += p;
      }
      l_run = l_run * alpha + lsum;
      m_run = mnew;
      sStat[wv][row] = alpha;
    }
    __syncthreads();

    // Rescale running accumulators
#pragma unroll
    for (int r = 0; r < 8; ++r) {
      float al = sStat[wv][r + 8 * hf];
      o0[r] *= al;
      o1[r] *= al;
    }

    // O += P @ V  (A-frag from per-wave LDS P tile; B-frag from staged V)
#pragma unroll
    for (int t = 0; t < 4; ++t) {
      int ka = 4 * t + 2 * hf;
      v2f a;
      a[0] = sP[wv][col][ka];
      a[1] = sP[wv][col][ka + 1];
      v2f b0, b1;
      b0[0] = sV[buf][ka][col];
      b0[1] = sV[buf][ka + 1][col];
      b1[0] = sV[buf][ka][16 + col];
      b1[1] = sV[buf][ka + 1][16 + col];
      o0 = wmma_f32_k4(a, b0, o0);
      o1 = wmma_f32_k4(a, b1, o1);
    }
    __syncthreads();   // protect buffers before next prefetch overwrites
  }

  if (lane < 16) sStat[wv][lane] = l_run;
  __syncthreads();

#pragma unroll
  for (int r = 0; r < 8; ++r) {
    int row = m0 + r + 8 * hf;
    float linv = 1.f / sStat[wv][r + 8 * hf];
    float it = (row < ANGLE_NUM) ? inten[row] : 0.f;
    O[(size_t)row * DMODEL + col] =
        o0[r] * linv + it * V[(size_t)row * DMODEL + col];
    O[(size_t)row * DMODEL + 16 + col] =
        o1[r] * linv + it * V[(size_t)row * DMODEL + 16 + col];
  }
}

// ---------------------------------------------------------------------------
// Row-wise LayerNorm over D=32
__global__ void ln_kernel(const float* __restrict__ in,
                          const float* __restrict__ g,
                          const float* __restrict__ b,
                          float* __restrict__ out) {
  int row = blockIdx.x * blockDim.x + threadIdx.x;
  if (row >= MP) return;
  const float* p = in + (size_t)row * DMODEL;
  float m = 0.f;
#pragma unroll
  for (int j = 0; j < DMODEL; ++j) m += p[j];
  m *= (1.f / DMODEL);
  float v = 0.f;
#pragma unroll
  for (int j = 0; j < DMODEL; ++j) {
    float d = p[j] - m;
    v += d * d;
  }
  v *= (1.f / DMODEL);
  float rs = rsqrtf(v + 1e-6f);
  float* q = out + (size_t)row * DMODEL;
#pragma unroll
  for (int j = 0; j < DMODEL; ++j) q[j] = (p[j] - m) * rs * g[j] + b[j];
}

// ---------------------------------------------------------------------------
// Final MLP: features[1024] = flat[272000] @ mlp_W[272000,1024] + mlp_b
__global__ void feat_init_kernel(const float* __restrict__ mlp_b,
                                 float* __restrict__ feat) {
  int j = blockIdx.x * blockDim.x + threadIdx.x;
  if (j < 1024) feat[j] = mlp_b[j];
}

__global__ void mlp_kernel(const float* __restrict__ flat,
                           const float* __restrict__ W,
                           float* __restrict__ feat) {
  __shared__ float sflat[MLP_CH];
  const int tid = threadIdx.x;  // 256 threads, 4 outputs each -> 1024 cols
  const int i0 = blockIdx.x * MLP_CH;
  const int NI = ANGLE_NUM * DMODEL;

  for (int t = tid; t < MLP_CH; t += 256) {
    int i = i0 + t;
    sflat[t] = (i < NI) ? flat[i] : 0.f;
  }
  __syncthreads();

  float4 acc = make_float4(0.f, 0.f, 0.f, 0.f);
  const int jb = tid * 4;
  int lim = NI - i0;
  if (lim > MLP_CH) lim = MLP_CH;
  for (int t = 0; t < lim; ++t) {
    __builtin_prefetch(W + (size_t)(i0 + t + 8) * 1024 + jb, 0, 1);
    float4 w4 = *(const float4*)(W + (size_t)(i0 + t) * 1024 + jb);
    float s = sflat[t];
    acc.x += s * w4.x;
    acc.y += s * w4.y;
    acc.z += s * w4.z;
    acc.w += s * w4.w;
  }
  atomicAdd(&feat[jb + 0], acc.x);
  atomicAdd(&feat[jb + 1], acc.y);
  atomicAdd(&feat[jb + 2], acc.z);
  atomicAdd(&feat[jb + 3], acc.w);
}

// sp[230] = features[1024] @ cls_W[1024,230] + cls_b
__global__ void cls_kernel(const float* __restrict__ feat,
                           const float* __restrict__ W,
                           const float* __restrict__ b,
                           float* __restrict__ sp) {
  __shared__ float sfeat[1024];
  for (int t = threadIdx.x; t < 1024; t += 256) sfeat[t] = feat[t];
  __syncthreads();
  int j = threadIdx.x;
  if (j >= 230) return;
  float acc = b[j];
  for (int k = 0; k < 1024; ++k) acc += sfeat[k] * W[k * 230 + j];
  sp[j] = acc;
}

// ---------------------------------------------------------------------------
extern "C" void kernel_launch(void* const* d_in, const int* in_sizes, int n_in,
                              void* d_out, int out_size, void* d_ws,
                              size_t ws_size, hipStream_t stream) {
  (void)in_sizes; (void)n_in; (void)out_size; (void)ws_size;
  const float* intensity = (const float*)d_in[0];
  const int*   angle     = (const int*)d_in[1];
  const float* embed     = (const float*)d_in[2];
  const float* Wq = (const float*)d_in[3];
  const float* bq = (const float*)d_in[4];
  const float* Wk = (const float*)d_in[5];
  const float* bk = (const float*)d_in[6];
  const float* Wv = (const float*)d_in[7];
  const float* bv = (const float*)d_in[8];
  const float* Wo = (const float*)d_in[9];
  const float* bo = (const float*)d_in[10];
  const float* ln1_g = (const float*)d_in[11];
  const float* ln1_b = (const float*)d_in[12];
  const float* W1 = (const float*)d_in[13];
  const float* b1 = (const float*)d_in[14];
  const float* W2 = (const float*)d_in[15];
  const float* b2 = (const float*)d_in[16];
  const float* ln2_g = (const float*)d_in[17];
  const float* ln2_b = (const float*)d_in[18];
  const float* mlp_W = (const float*)d_in[19];
  const float* mlp_b = (const float*)d_in[20];
  const float* cls_W = (const float*)d_in[21];
  const float* cls_b = (const float*)d_in[22];

  float* ws = (float*)d_ws;
  const size_t seg = (size_t)MP * DMODEL;
  float* x  = ws + 0 * seg;
  float* q  = ws + 1 * seg;
  float* k  = ws + 2 * seg;
  float* v  = ws + 3 * seg;
  float* ao = ws + 4 * seg;
  float* t1 = ws + 5 * seg;
  float* h  = ws + 6 * seg;
  float* t2 = ws + 7 * seg;
  float* f1 = ws + 8 * seg;  // [MP, 64] -> 2 segs

  float* feat = (float*)d_out;   // 1024
  float* sp   = feat + 1024;     // 230

  embed_kernel<<<(MP * DMODEL + 255) / 256, 256, 0, stream>>>(intensity, angle,
                                                              embed, x);

  dim3 g2(MP / 64, 2);  // N = 32
  dim3 g4(MP / 64, 4);  // N = 64
  for (int i = 0; i < NLAYERS; ++i) {
    const float* Wqi = Wq + (size_t)i * DMODEL * DMODEL;
    const float* Wki = Wk + (size_t)i * DMODEL * DMODEL;
    const float* Wvi = Wv + (size_t)i * DMODEL * DMODEL;
    const float* Woi = Wo + (size_t)i * DMODEL * DMODEL;
    const float* W1i = W1 + (size_t)i * DMODEL * DFF;
    const float* W2i = W2 + (size_t)i * DFF * DMODEL;

    gemm16_kernel<<<g2, 128, 0, stream>>>(x, Wqi, bq + i * DMODEL, nullptr, q,
                                          DMODEL, DMODEL, 0);
    gemm16_kernel<<<g2, 128, 0, stream>>>(x, Wki, bk + i * DMODEL, nullptr, k,
                                          DMODEL, DMODEL, 0);
    gemm16_kernel<<<g2, 128, 0, stream>>>(x, Wvi, bv + i * DMODEL, nullptr, v,
                                          DMODEL, DMODEL, 0);
    attn_kernel<<<MP / 64, 128, 0, stream>>>(q, k, v, intensity, ao);
    gemm16_kernel<<<g2, 128, 0, stream>>>(ao, Woi, bo + i * DMODEL, x, t1,
                                          DMODEL, DMODEL, 0);
    ln_kernel<<<(MP + 255) / 256, 256, 0, stream>>>(t1, ln1_g + i * DMODEL,
                                                    ln1_b + i * DMODEL, h);
    gemm16_kernel<<<g4, 128, 0, stream>>>(h, W1i, b1 + i * DFF, nullptr, f1,
                                          DMODEL, DFF, 1);
    gemm16_kernel<<<g2, 128, 0, stream>>>(f1, W2i, b2 + i * DMODEL, h, t2, DFF,
                                          DMODEL, 0);
    ln_kernel<<<(MP + 255) / 256, 256, 0, stream>>>(t2, ln2_g + i * DMODEL,
                                                    ln2_b + i * DMODEL, x);
  }

  feat_init_kernel<<<4, 256, 0, stream>>>(mlp_b, feat);
  mlp_kernel<<<(ANGLE_NUM * DMODEL + MLP_CH - 1) / MLP_CH, 256, 0, stream>>>(
      x, mlp_W, feat);
  cls_kernel<<<1, 256, 0, stream>>>(feat, cls_W, cls_b, sp);
}